// GraphAttention_6511170421123
// MI455X (gfx1250) — compile-verified
//
#include <hip/hip_runtime.h>
#include <hip/hip_bf16.h>
#include <math.h>

// ---------------------------------------------------------------------------
// GAT layer for gfx1250 (MI455X), fp32. WMMA f32 16x16x4 for both GEMMs with
// LDS-staged weights (conflict-free ds_load_b64 fragments) + pipelined A
// loads; fp32 global atomics (L2-resident) for segment softmax + aggregate.
// ---------------------------------------------------------------------------

#define GN   50000
#define GE   800000
#define GH   4
#define GD   32
#define GIN  128
#define GEF  64
#define GHD  128          // H*D
#define GALPHA 0.2f
#define GEPS   1e-5f

typedef __attribute__((ext_vector_type(2))) float v2f;
typedef __attribute__((ext_vector_type(8))) float v8f;

__device__ __forceinline__ float leakyf(float v) {
    return v >= 0.0f ? v : GALPHA * v;
}

__device__ __forceinline__ void atomicMaxF(float* addr, float v) {
    // lowers to global_atomic_max_num_f32 (agent scope)
    __hip_atomic_fetch_max(addr, v, __ATOMIC_RELAXED, __HIP_MEMORY_SCOPE_AGENT);
}

// ---------------------------------------------------------------------------
// init: agg=0, z=0, amax=-inf, bn sums=0
// ---------------------------------------------------------------------------
__global__ void k_init(float* __restrict__ agg, float* __restrict__ z,
                       float* __restrict__ amax, float* __restrict__ sums) {
    int i = blockIdx.x * blockDim.x + threadIdx.x;
    if (i < GN * GHD) agg[i] = 0.0f;
    if (i < GN * GH)  { z[i] = 0.0f; amax[i] = -3.0e38f; }
    if (i < 2 * GHD)  sums[i] = 0.0f;
}

// ---------------------------------------------------------------------------
// ft = x @ W^T  : [N,128] = [N,128] x [128,128]^T
// W staged in LDS as (k-pair, col) float2 -> ds_load_b64 fragments, lanes
// stride 8B (bank-conflict-free). A fragments software-pipelined from global.
// One wave -> 16 rows x 128 cols, 8 WMMA accumulators.
// ---------------------------------------------------------------------------
__global__ void k_nodeproj(const float* __restrict__ x,
                           const float* __restrict__ W,
                           float* __restrict__ ft) {
    __shared__ float2 sW[64 * 128];      // (kp, col): {W[col][2kp], W[col][2kp+1]}  64KB
    const int tid = threadIdx.x;

    // stage W: coalesced global float2 reads, scattered LDS writes (one-time)
    for (int i = tid; i < 64 * 128; i += 256) {
        const int col = i >> 6;          // 0..127
        const int kp  = i & 63;          // 0..63
        float2 v;
        v.x = W[(size_t)col * GIN + 2 * kp];
        v.y = W[(size_t)col * GIN + 2 * kp + 1];
        sW[kp * 128 + col] = v;
    }
    __syncthreads();

    const int lane = tid & 31;
    const int wv   = tid >> 5;
    const int tile = blockIdx.x * 8 + wv;            // 16-row tile id
    if (tile < GN / 16) {                            // wave-uniform
        const int m0   = tile * 16;
        const int half = lane >> 4;                  // 0/1: K pair select
        const int l16  = lane & 15;

        v8f acc[8];
        #pragma unroll
        for (int t = 0; t < 8; ++t) acc[t] = (v8f){0,0,0,0,0,0,0,0};

        const float* xr = x + (size_t)(m0 + l16) * GIN;
        __builtin_prefetch(xr, 0, 3);                // global_prefetch_b8
        const int kb = 2 * half;

        v2f af; af.x = xr[kb]; af.y = xr[kb + 1];    // A frag for k = 0
        for (int k = 0; k < GIN; k += 4) {
            v2f afn = af;                            // pipeline next A frag
            if (k + 4 < GIN) {
                afn.x = xr[k + 4 + kb];
                afn.y = xr[k + 4 + kb + 1];
            }
            const int kp = (k >> 1) + half;          // (k + 2*half)/2
            v2f bf[8];
            #pragma unroll
            for (int t = 0; t < 8; ++t) {            // 8 ds_load_b64, one wait
                float2 p = sW[kp * 128 + t * 16 + l16];
                bf[t].x = p.x; bf[t].y = p.y;
            }
            #pragma unroll
            for (int t = 0; t < 8; ++t) {
                acc[t] = __builtin_amdgcn_wmma_f32_16x16x4_f32(
                    false, af, false, bf[t], (short)0, acc[t], false, false);
            }
            af = afn;
        }
        // D layout: VGPR v -> row m0 + v + 8*half, col = t*16 + l16
        #pragma unroll
        for (int t = 0; t < 8; ++t) {
            #pragma unroll
            for (int v = 0; v < 8; ++v) {
                ft[(size_t)(m0 + v + 8 * half) * GHD + t * 16 + l16] = acc[t][v];
            }
        }
    }
}

// ---------------------------------------------------------------------------
// a1[n,h] = <ft[n,h,:], attn_l[h,:]> ; a2 likewise
// ---------------------------------------------------------------------------
__global__ void k_scores(const float* __restrict__ ft,
                         const float* __restrict__ attn_l,
                         const float* __restrict__ attn_r,
                         float* __restrict__ a1, float* __restrict__ a2) {
    int i = blockIdx.x * blockDim.x + threadIdx.x;   // (n,h) flat
    if (i >= GN * GH) return;
    const int h = i & (GH - 1);
    const int n = i >> 2;
    const float* f  = ft + (size_t)n * GHD + h * GD;
    const float* al = attn_l + h * GD;
    const float* ar = attn_r + h * GD;
    float s1 = 0.0f, s2 = 0.0f;
    #pragma unroll
    for (int d = 0; d < GD; ++d) { s1 += f[d] * al[d]; s2 += f[d] * ar[d]; }
    a1[i] = s1; a2[i] = s2;
}

// ---------------------------------------------------------------------------
// edge MLP + raw scores + segment-max (fused)
// wave -> 16 edges; WMMA: [16x64]x[64x32] hidden (We1 staged in LDS),
// leaky -> per-wave LDS tile (s_wait_dscnt handoff), then per-lane 2 outputs
// of second linear [32->4 heads], add a1/a2, leaky, raw score + atomicMax.
// ---------------------------------------------------------------------------
__global__ void k_edge(const float* __restrict__ edge_x,
                       const float* __restrict__ We1,
                       const float* __restrict__ be1,
                       const float* __restrict__ We2,
                       const float* __restrict__ be2,
                       const float* __restrict__ a1,
                       const float* __restrict__ a2,
                       const int* __restrict__ src,
                       const int* __restrict__ dst,
                       float* __restrict__ a_out,
                       float* __restrict__ amax) {
    __shared__ float2 sW1[32 * 32];      // (kp, col): {We1[col][2kp], We1[col][2kp+1]} 8KB
    __shared__ float  tileL[8][16 * 32]; // per-wave hidden tile, 16KB
    const int tid = threadIdx.x;

    // stage We1 (one-time)
    for (int i = tid; i < 32 * 32; i += 256) {
        const int col = i >> 5;          // 0..31
        const int kp  = i & 31;          // 0..31
        float2 v;
        v.x = We1[(size_t)col * GEF + 2 * kp];
        v.y = We1[(size_t)col * GEF + 2 * kp + 1];
        sW1[kp * 32 + col] = v;
    }
    __syncthreads();

    const int lane = tid & 31;
    const int wv   = tid >> 5;
    const int tile = blockIdx.x * 8 + wv;            // E/16 = 50000 tiles exact
    const int e0   = tile * 16;
    const int half = lane >> 4;
    const int l16  = lane & 15;
    const int kb   = 2 * half;

    v8f acc0 = (v8f){0,0,0,0,0,0,0,0};
    v8f acc1 = (v8f){0,0,0,0,0,0,0,0};
    const float* exr = edge_x + (size_t)(e0 + l16) * GEF;
    __builtin_prefetch(exr, 0, 3);                   // global_prefetch_b8

    v2f af; af.x = exr[kb]; af.y = exr[kb + 1];
    for (int k = 0; k < GEF; k += 4) {
        v2f afn = af;
        if (k + 4 < GEF) {
            afn.x = exr[k + 4 + kb];
            afn.y = exr[k + 4 + kb + 1];
        }
        const int kp = (k >> 1) + half;
        float2 p0 = sW1[kp * 32 + l16];
        float2 p1 = sW1[kp * 32 + 16 + l16];
        v2f b0; b0.x = p0.x; b0.y = p0.y;
        v2f b1; b1.x = p1.x; b1.y = p1.y;
        acc0 = __builtin_amdgcn_wmma_f32_16x16x4_f32(
            false, af, false, b0, (short)0, acc0, false, false);
        acc1 = __builtin_amdgcn_wmma_f32_16x16x4_f32(
            false, af, false, b1, (short)0, acc1, false, false);
        af = afn;
    }

    // bias + leaky -> per-wave LDS tile (row = v + 8*half, col = t*16 + l16)
    float* L = tileL[wv];
    const float bl0 = be1[l16], bl1 = be1[16 + l16];
    #pragma unroll
    for (int v = 0; v < 8; ++v) {
        const int row = v + 8 * half;
        L[row * 32 + l16]      = leakyf(acc0[v] + bl0);
        L[row * 32 + 16 + l16] = leakyf(acc1[v] + bl1);
    }
    // wave-private tile: DS ops are in-order per wave, just drain DScnt
    asm volatile("s_wait_dscnt 0x0" ::: "memory");

    // second linear: 64 outputs (16 rows x 4 heads) -> 2 per lane
    #pragma unroll
    for (int jj = 0; jj < 2; ++jj) {
        const int j = lane * 2 + jj;                 // 0..63
        const int r = j >> 2;                        // row in tile
        const int h = j & 3;                         // head
        const float* hr = L + r * 32;
        const float* w2 = We2 + h * GD;
        float s = be2[h];
        #pragma unroll
        for (int d = 0; d < GD; ++d) s += hr[d] * w2[d];
        const int e  = e0 + r;
        const int sn = src[e];
        const int dn = dst[e];
        s += a1[sn * GH + h] + a2[dn * GH + h];
        s = leakyf(s);
        a_out[(size_t)e * GH + h] = s;
        atomicMaxF(&amax[dn * GH + h], s);
    }
}

// ---------------------------------------------------------------------------
// exp-softmax numerator + z accumulation + weighted scatter-aggregate
// one wave per edge: 32 lanes x float4 = 128 channels
// ---------------------------------------------------------------------------
__global__ void k_agg(const float* __restrict__ a,
                      const float* __restrict__ amax,
                      const int* __restrict__ src,
                      const int* __restrict__ dst,
                      const float* __restrict__ ft,
                      float* __restrict__ z,
                      float* __restrict__ agg) {
    const int lane = threadIdx.x & 31;
    const int wv   = threadIdx.x >> 5;
    const int e    = blockIdx.x * 8 + wv;            // E/8 = 100000 blocks exact
    const int sn = src[e];
    const int dn = dst[e];
    const float p0 = __expf(a[(size_t)e * GH + 0] - amax[dn * GH + 0]);
    const float p1 = __expf(a[(size_t)e * GH + 1] - amax[dn * GH + 1]);
    const float p2 = __expf(a[(size_t)e * GH + 2] - amax[dn * GH + 2]);
    const float p3 = __expf(a[(size_t)e * GH + 3] - amax[dn * GH + 3]);
    if (lane == 0) {
        atomicAdd(&z[dn * GH + 0], p0);
        atomicAdd(&z[dn * GH + 1], p1);
        atomicAdd(&z[dn * GH + 2], p2);
        atomicAdd(&z[dn * GH + 3], p3);
    }
    const int c = lane * 4;                          // 4 channels, same head
    const int h = c >> 5;
    const float ph = (h == 0) ? p0 : (h == 1) ? p1 : (h == 2) ? p2 : p3;
    const float4 f4 = *(const float4*)(ft + (size_t)sn * GHD + c);
    float* ag = agg + (size_t)dn * GHD + c;
    atomicAdd(&ag[0], ph * f4.x);
    atomicAdd(&ag[1], ph * f4.y);
    atomicAdd(&ag[2], ph * f4.z);
    atomicAdd(&ag[3], ph * f4.w);
}

// ---------------------------------------------------------------------------
// pre = agg / z -> d_out ; per-channel sum & sumsq via LDS then global atomics
// block = 256 threads, handles 32 rows x 128 channels
// ---------------------------------------------------------------------------
__global__ void k_prebn(const float* __restrict__ agg,
                        const float* __restrict__ z,
                        float* __restrict__ out,
                        float* __restrict__ sums) {     // [0:128)=sum [128:256)=sumsq
    __shared__ float ls[GHD], lq[GHD];
    const int tid  = threadIdx.x;
    const int c    = tid & (GHD - 1);
    const int rsub = tid >> 7;                        // 0/1
    if (tid < GHD) { ls[tid] = 0.0f; lq[tid] = 0.0f; }
    __syncthreads();

    float s = 0.0f, sq = 0.0f;
    #pragma unroll
    for (int i = 0; i < 16; ++i) {
        const int r = blockIdx.x * 32 + rsub * 16 + i;
        if (r < GN) {
            const float pre = agg[(size_t)r * GHD + c] / z[r * GH + (c >> 5)];
            out[(size_t)r * GHD + c] = pre;
            s += pre; sq += pre * pre;
        }
    }
    atomicAdd(&ls[c], s);
    atomicAdd(&lq[c], sq);
    __syncthreads();
    if (tid < GHD) {
        atomicAdd(&sums[tid],       ls[tid]);
        atomicAdd(&sums[GHD + tid], lq[tid]);
    }
}

// ---------------------------------------------------------------------------
// batchnorm finalize (in place on d_out)
// ---------------------------------------------------------------------------
__global__ void k_bn(float* __restrict__ out,
                     const float* __restrict__ sums,
                     const float* __restrict__ gamma,
                     const float* __restrict__ beta) {
    const int i = blockIdx.x * blockDim.x + threadIdx.x;
    if (i >= GN * GHD) return;
    const int c = i & (GHD - 1);
    const float inv_n = 1.0f / (float)GN;
    const float mean = sums[c] * inv_n;
    const float var  = sums[GHD + c] * inv_n - mean * mean;
    out[i] = (out[i] - mean) * rsqrtf(var + GEPS) * gamma[c] + beta[c];
}

// ---------------------------------------------------------------------------
extern "C" void kernel_launch(void* const* d_in, const int* in_sizes, int n_in,
                              void* d_out, int out_size, void* d_ws, size_t ws_size,
                              hipStream_t stream) {
    (void)in_sizes; (void)n_in; (void)out_size; (void)ws_size;
    const float* x      = (const float*)d_in[0];
    const float* edge_x = (const float*)d_in[1];
    const int*   src    = (const int*)  d_in[2];
    const int*   dst    = (const int*)  d_in[3];
    const float* W      = (const float*)d_in[4];
    const float* attn_l = (const float*)d_in[5];
    const float* attn_r = (const float*)d_in[6];
    const float* We1    = (const float*)d_in[7];
    const float* be1    = (const float*)d_in[8];
    const float* We2    = (const float*)d_in[9];
    const float* be2    = (const float*)d_in[10];
    const float* gamma  = (const float*)d_in[11];
    const float* beta   = (const float*)d_in[12];
    float* out = (float*)d_out;

    // workspace layout (floats)
    float* ws   = (float*)d_ws;
    float* ft   = ws;                          size_t off = (size_t)GN * GHD;
    float* a1   = ws + off;                    off += (size_t)GN * GH;
    float* a2   = ws + off;                    off += (size_t)GN * GH;
    float* a    = ws + off;                    off += (size_t)GE * GH;
    float* amax = ws + off;                    off += (size_t)GN * GH;
    float* zbuf = ws + off;                    off += (size_t)GN * GH;
    float* agg  = ws + off;                    off += (size_t)GN * GHD;
    float* sums = ws + off;                    // 2*128 floats

    const int BT = 256;

    k_init<<<(GN * GHD + BT - 1) / BT, BT, 0, stream>>>(agg, zbuf, amax, sums);

    // node projection: 3125 row-tiles, 8 waves/block
    k_nodeproj<<<(GN / 16 + 7) / 8, BT, 0, stream>>>(x, W, ft);

    k_scores<<<(GN * GH + BT - 1) / BT, BT, 0, stream>>>(ft, attn_l, attn_r, a1, a2);

    // edge MLP + raw scores + segment max: 50000 tiles / 8 waves = 6250 blocks
    k_edge<<<GE / 16 / 8, BT, 0, stream>>>(edge_x, We1, be1, We2, be2,
                                           a1, a2, src, dst, a, amax);

    // softmax numerators + z + scatter-aggregate: wave per edge
    k_agg<<<GE / 8, BT, 0, stream>>>(a, amax, src, dst, ft, zbuf, agg);

    // normalize by z + BN statistics
    k_prebn<<<(GN + 31) / 32, BT, 0, stream>>>(agg, zbuf, out, sums);

    // BN finalize
    k_bn<<<(GN * GHD + BT - 1) / BT, BT, 0, stream>>>(out, sums, gamma, beta);
}